// GATLayer_86363202388377
// MI455X (gfx1250) — compile-verified
//
#include <hip/hip_runtime.h>
#include <stdint.h>

#define NN 4096      // nodes
#define NE 131072    // edges
#define DD 256       // feature dim
#define NH 8         // heads
#define LN_EPS 1e-5f
#define MAXDEG 256   // per-row bucket bound held in LDS (avg in-degree = 32)

typedef __attribute__((ext_vector_type(2))) float v2f;
typedef __attribute__((ext_vector_type(8))) float v8f;

// ---------------------------------------------------------------------------
// Zero the accumulators that are atomically built each call.
__global__ void __launch_bounds__(256) zero_kernel(float* Dsum, int* cnt, float* colsum) {
    int t = blockIdx.x * 256 + threadIdx.x;
    if (t < NN * NH) Dsum[t] = 0.0f;
    if (t < NN)      cnt[t]  = 0;
    if (t < DD)      colsum[t] = 0.0f;
}

// ---------------------------------------------------------------------------
// Fused Q/K/V GEMM: out = x @ W^T + b, via V_WMMA_F32_16X16X4_F32.
// One wave computes one 16x16 tile; block = 8 waves (8 column tiles),
// grid = (row_tiles=256, col_tile_groups=2, matrix=3).
__global__ void __launch_bounds__(256) qkv_wmma_kernel(
    const float* __restrict__ x,
    const float* __restrict__ Wq, const float* __restrict__ bq,
    const float* __restrict__ Wk, const float* __restrict__ bk,
    const float* __restrict__ Wv, const float* __restrict__ bv,
    float* __restrict__ Q, float* __restrict__ K, float* __restrict__ V)
{
    const int mat = blockIdx.z;
    const float* __restrict__ W    = (mat == 0) ? Wq : (mat == 1) ? Wk : Wv;
    const float* __restrict__ bias = (mat == 0) ? bq : (mat == 1) ? bk : bv;
    float* __restrict__ out        = (mat == 0) ? Q  : (mat == 1) ? K  : V;

    const int row0 = blockIdx.x * 16;
    const int col0 = (blockIdx.y * 8 + threadIdx.y) * 16;
    const int lane = threadIdx.x;          // 0..31 (wave32)
    const int m    = lane & 15;            // row within tile (A) / col within tile (B)
    const int kb   = (lane >> 4) << 1;     // K sub-offset: 0 or 2

    // A: lane holds x[row0+m][k+kb .. k+kb+1]
    // B: lane holds W[col0+m][k+kb .. k+kb+1]  (B = W^T tile, K-major per ISA layout)
    const float* __restrict__ xrow = x + (size_t)(row0 + m) * DD + kb;
    const float* __restrict__ wrow = W + (size_t)(col0 + m) * DD + kb;

    v8f acc = {0.f, 0.f, 0.f, 0.f, 0.f, 0.f, 0.f, 0.f};
    for (int k = 0; k < DD; k += 4) {
        __builtin_prefetch(xrow + k + 64, 0, 0);   // global_prefetch_b8
        v2f a = *(const v2f*)(xrow + k);
        v2f b = *(const v2f*)(wrow + k);
        acc = __builtin_amdgcn_wmma_f32_16x16x4_f32(
            /*neg_a=*/false, a, /*neg_b=*/false, b,
            /*c_mod=*/(short)0, acc, /*reuse_a=*/false, /*reuse_b=*/false);
    }

    // C/D layout: VGPR v -> row = row0 + 8*(lane>>4) + v, col = col0 + (lane&15)
    const int orow = row0 + ((lane >> 4) << 3);
    const int ocol = col0 + m;
    const float bb = bias[ocol];
#pragma unroll
    for (int v = 0; v < 8; ++v)
        out[(size_t)(orow + v) * DD + ocol] = acc[v] + bb;
}

// ---------------------------------------------------------------------------
// colsum[d] = sum_j V[j][d]  (block b covers 16 rows; coalesced per-column)
__global__ void __launch_bounds__(256) colsum_kernel(const float* __restrict__ V,
                                                     float* __restrict__ colsum) {
    int d = threadIdx.x;
    int r0 = blockIdx.x * 16;
    float s = 0.f;
#pragma unroll
    for (int j = 0; j < 16; ++j) s += V[(size_t)(r0 + j) * DD + d];
    atomicAdd(&colsum[d], s);
}

// ---------------------------------------------------------------------------
// Per (edge, head): score -> leaky_relu -> *ew -> exp; accumulate denominators.
// attn (512 f32) is staged into LDS via CDNA5 async global->LDS copy.
__global__ void __launch_bounds__(256) edge_score_kernel(
    const long long* __restrict__ ei, const float* __restrict__ ew,
    const float* __restrict__ Q, const float* __restrict__ K,
    const float* __restrict__ attn,
    float* __restrict__ Eex, float* __restrict__ Dsum, int* __restrict__ cnt)
{
    __shared__ float sa[2 * DD];
    int t = threadIdx.x;
    {
        // LDS byte offset = low 32 bits of the generic shared address
        unsigned lds0 = (unsigned)(uintptr_t)(&sa[t]);
        unsigned lds1 = (unsigned)(uintptr_t)(&sa[t + 256]);
        unsigned long long g0 = (unsigned long long)(uintptr_t)(attn + t);
        unsigned long long g1 = (unsigned long long)(uintptr_t)(attn + t + 256);
        asm volatile("global_load_async_to_lds_b32 %0, %1, off"
                     :: "v"(lds0), "v"(g0) : "memory");
        asm volatile("global_load_async_to_lds_b32 %0, %1, off"
                     :: "v"(lds1), "v"(g1) : "memory");
        asm volatile("s_wait_asynccnt 0x0" ::: "memory");
    }
    __syncthreads();

    int gid = blockIdx.x * 256 + t;        // 0 .. NE*NH-1
    int e = gid >> 3;
    int h = gid & 7;
    long long s = ei[e];
    long long d = ei[NE + e];

    const float* __restrict__ q = Q + (size_t)s * DD + h * 32;
    const float* __restrict__ k = K + (size_t)d * DD + h * 32;
    const float* __restrict__ a = sa + h * 64;

    float acc = 0.f;
#pragma unroll
    for (int i = 0; i < 32; ++i) acc += q[i] * a[i] + k[i] * a[32 + i];
    acc = (acc < 0.f) ? 0.2f * acc : acc;  // leaky_relu(0.2)
    acc *= ew[e];
    float ex = expf(acc);

    Eex[gid] = ex;
    atomicAdd(&Dsum[(size_t)d * NH + h], ex);
    if (h == 0) atomicAdd(&cnt[(int)d], 1);
}

// ---------------------------------------------------------------------------
// invD[i][k] = 1 / ((N - indeg(i)) + Dsum[i][k]);  c[i] = sum_k invD[i][k]
__global__ void __launch_bounds__(256) node_finalize_kernel(
    const float* __restrict__ Dsum, const int* __restrict__ cnt,
    float* __restrict__ invD, float* __restrict__ c)
{
    int i = blockIdx.x * 256 + threadIdx.x;
    if (i >= NN) return;
    float base = (float)(NN - cnt[i]);
    float ci = 0.f;
#pragma unroll
    for (int k = 0; k < NH; ++k) {
        float iv = 1.0f / (base + Dsum[i * NH + k]);
        invD[i * NH + k] = iv;
        ci += iv;
    }
    c[i] = ci;
}

// ---------------------------------------------------------------------------
// Exclusive prefix sum of in-degrees -> CSR row starts (+ working cursor).
// Single block: thread t owns 16 consecutive nodes; Hillis-Steele on partials.
__global__ void __launch_bounds__(256) scan_kernel(const int* __restrict__ cnt,
                                                   int* __restrict__ rowstart,
                                                   int* __restrict__ cursor)
{
    __shared__ int sp[256];
    int t = threadIdx.x;
    int base = t * 16;
    int local[16];
    int p = 0;
#pragma unroll
    for (int j = 0; j < 16; ++j) { local[j] = cnt[base + j]; p += local[j]; }
    sp[t] = p;
    __syncthreads();
    for (int off = 1; off < 256; off <<= 1) {
        int v = (t >= off) ? sp[t - off] : 0;
        __syncthreads();
        sp[t] += v;
        __syncthreads();
    }
    int run = sp[t] - p;   // exclusive prefix
#pragma unroll
    for (int j = 0; j < 16; ++j) {
        rowstart[base + j] = run;
        cursor[base + j]   = run;
        run += local[j];
    }
}

// ---------------------------------------------------------------------------
// Per-edge weight w[e] = sum_k Eex[e][k]*invD[dst][k] - c[dst], and bucket the
// edge id into its destination row (CSR fill).
__global__ void __launch_bounds__(256) edge_w_fill_kernel(
    const long long* __restrict__ ei, const float* __restrict__ Eex,
    const float* __restrict__ invD, const float* __restrict__ c,
    float* __restrict__ wE, int* __restrict__ cursor, int* __restrict__ beid)
{
    int e = blockIdx.x * 256 + threadIdx.x;
    if (e >= NE) return;
    int d = (int)ei[NE + e];
    float be = 0.f;
#pragma unroll
    for (int k = 0; k < NH; ++k) be += Eex[e * NH + k] * invD[d * NH + k];
    wE[e] = be - c[d];
    int pos = atomicAdd(&cursor[d], 1);
    beid[pos] = e;
}

// ---------------------------------------------------------------------------
// Fused gather + residual + LayerNorm. One block per node row; thread = dim.
// Row's edge ids are sorted in LDS so the FP accumulation order is
// deterministic across replays.
__global__ void __launch_bounds__(256) gather_ln_kernel(
    const long long* __restrict__ ei, const int* __restrict__ cnt,
    const int* __restrict__ rowstart, const int* __restrict__ beid,
    const float* __restrict__ wE, const float* __restrict__ V,
    const float* __restrict__ cN, const float* __restrict__ colsum,
    const float* __restrict__ x, const float* __restrict__ gamma,
    const float* __restrict__ beta, float* __restrict__ out)
{
    __shared__ int   sh[MAXDEG];
    __shared__ float sb[256];
    int i = blockIdx.x, t = threadIdx.x;
    int deg = cnt[i], start = rowstart[i];
    bool inlds = (deg <= MAXDEG);
    for (int j = t; j < deg && j < MAXDEG; j += 256) sh[j] = beid[start + j];
    __syncthreads();
    if (t == 0 && inlds) {                 // insertion sort by edge id
        for (int a = 1; a < deg; ++a) {
            int key = sh[a], b = a - 1;
            while (b >= 0 && sh[b] > key) { sh[b + 1] = sh[b]; --b; }
            sh[b + 1] = key;
        }
    }
    __syncthreads();

    float acc = cN[i] * colsum[t];         // background term of the softmax sum
    for (int j = 0; j < deg; ++j) {
        int e = inlds ? sh[j] : beid[start + j];
        int s = (int)ei[e];
        acc += wE[e] * V[(size_t)s * DD + t];
    }

    float h = acc + x[(size_t)i * DD + t]; // residual
    sb[t] = h;
    __syncthreads();
    for (int s = 128; s > 0; s >>= 1) { if (t < s) sb[t] += sb[t + s]; __syncthreads(); }
    float mu = sb[0] * (1.0f / 256.0f);
    __syncthreads();
    float dm = h - mu;
    sb[t] = dm * dm;
    __syncthreads();
    for (int s = 128; s > 0; s >>= 1) { if (t < s) sb[t] += sb[t + s]; __syncthreads(); }
    float var = sb[0] * (1.0f / 256.0f);
    out[(size_t)i * DD + t] = dm * rsqrtf(var + LN_EPS) * gamma[t] + beta[t];
}

// ---------------------------------------------------------------------------
extern "C" void kernel_launch(void* const* d_in, const int* in_sizes, int n_in,
                              void* d_out, int out_size, void* d_ws, size_t ws_size,
                              hipStream_t stream) {
    const float*     x     = (const float*)d_in[0];
    const long long* ei    = (const long long*)d_in[1];   // int64 (2, NE)
    const float*     ew    = (const float*)d_in[2];
    const float*     Wq    = (const float*)d_in[3];
    const float*     bq    = (const float*)d_in[4];
    const float*     Wk    = (const float*)d_in[5];
    const float*     bk    = (const float*)d_in[6];
    const float*     Wv    = (const float*)d_in[7];
    const float*     bv    = (const float*)d_in[8];
    const float*     attn  = (const float*)d_in[9];
    const float*     gamma = (const float*)d_in[10];
    const float*     beta  = (const float*)d_in[11];
    float* out = (float*)d_out;

    // Workspace layout
    float* ws = (float*)d_ws;
    float* Q      = ws;                          // NN*DD
    float* K      = Q + (size_t)NN * DD;         // NN*DD
    float* V      = K + (size_t)NN * DD;         // NN*DD
    float* Eex    = V + (size_t)NN * DD;         // NE*NH
    float* wE     = Eex + (size_t)NE * NH;       // NE
    float* Dsum   = wE + (size_t)NE;             // NN*NH
    float* invD   = Dsum + (size_t)NN * NH;      // NN*NH
    float* cN     = invD + (size_t)NN * NH;      // NN
    float* colsum = cN + (size_t)NN;             // DD
    int*   cnt      = (int*)(colsum + DD);       // NN
    int*   rowstart = cnt + NN;                  // NN
    int*   cursor   = rowstart + NN;             // NN
    int*   beid     = cursor + NN;               // NE

    // 1) zero atomically-built accumulators (deterministic across replays)
    zero_kernel<<<dim3((NN * NH + 255) / 256), dim3(256), 0, stream>>>(Dsum, cnt, colsum);

    // 2) Q/K/V = x @ W^T + b  (WMMA f32 16x16x4)
    qkv_wmma_kernel<<<dim3(NN / 16, 2, 3), dim3(32, 8), 0, stream>>>(
        x, Wq, bq, Wk, bk, Wv, bv, Q, K, V);

    // 3) column sums of V
    colsum_kernel<<<dim3(NN / 16), dim3(256), 0, stream>>>(V, colsum);

    // 4) per-(edge,head) exp-scores + denominators + in-degree
    edge_score_kernel<<<dim3(NE * NH / 256), dim3(256), 0, stream>>>(
        ei, ew, Q, K, attn, Eex, Dsum, cnt);

    // 5) per-node 1/D and c_i
    node_finalize_kernel<<<dim3(NN / 256), dim3(256), 0, stream>>>(Dsum, cnt, invD, cN);

    // 6) CSR row starts from in-degrees
    scan_kernel<<<dim3(1), dim3(256), 0, stream>>>(cnt, rowstart, cursor);

    // 7) per-edge aggregate weight + CSR bucket fill
    edge_w_fill_kernel<<<dim3(NE / 256), dim3(256), 0, stream>>>(
        ei, Eex, invD, cN, wE, cursor, beid);

    // 8) atomic-free gather + residual + LayerNorm
    gather_ln_kernel<<<dim3(NN), dim3(256), 0, stream>>>(
        ei, cnt, rowstart, beid, wE, V, cN, colsum, x, gamma, beta, out);
}